// BasicBlock_28046136443119
// MI455X (gfx1250) — compile-verified
//
#include <hip/hip_runtime.h>

// ---------------------------------------------------------------------------
// BasicBlock (fp32) for MI455X / gfx1250:
//   conv1 3x3 s2 128->256  -> percentiles(99) -> clamp01 -> conv2 3x3 s1 +
//   1x1 s2 residual        -> percentiles(99) -> clamp01
// Convs: implicit GEMM on V_WMMA_F32_16X16X4_F32 (fp32 matrix pipe, exact
// numerics for the order statistics). Weight tiles staged to LDS with
// GLOBAL_LOAD_ASYNC_TO_LDS_B128 (ASYNCcnt-tracked). Percentiles: exact k-th
// values via 3-level radix histogram descent (12/12/8 bits).
// ---------------------------------------------------------------------------

typedef __attribute__((ext_vector_type(2))) float v2f;
typedef __attribute__((ext_vector_type(8))) float v8f;
typedef __attribute__((ext_vector_type(4))) int   v4i;

#define NOUT_LL (64LL * 256 * 28 * 28)   // 12,845,056
#define BM 128
#define BN 64
#define BK 16
#define NTHREADS 256

#if __has_builtin(__builtin_amdgcn_global_load_async_to_lds_b128) && \
    __has_builtin(__builtin_amdgcn_s_wait_asynccnt)
#define HAVE_ASYNC_LDS 1
typedef __attribute__((address_space(1))) v4i* gv4i_ptr;  // global int4*
typedef __attribute__((address_space(3))) v4i* lv4i_ptr;  // LDS int4*
#else
#define HAVE_ASYNC_LDS 0
#endif

// ---------------------------------------------------------------------------
// Implicit-GEMM conv. MODE 0: conv1 (K=1152). MODE 1: conv2 + fused residual
// (K = 2304 + 128; tail K-range is the 1x1/s2 identity path).
// A = im2col(x) [M=50176, K], B = weights [K, N=256], C written NCHW.
// ---------------------------------------------------------------------------
template <int MODE>
__global__ __launch_bounds__(NTHREADS) void conv_gemm(
    const float* __restrict__ x,    // [64,128,56,56]
    const float* __restrict__ w,    // w1 [256,1152] or w2 [256,2304]
    const float* __restrict__ wid,  // w_id [256,128] (MODE 1)
    const float* __restrict__ act,  // raw conv1 out [64,256,28,28] (MODE 1)
    float* __restrict__ out)        // raw output [64,256,28,28]
{
    __shared__ float As[BM][BK + 2];
    __shared__ float Bs[BN][BK + 4];  // row stride 80 B: 16B-aligned b128 slots

    const int tid  = threadIdx.x;
    const int m0   = blockIdx.y * BM;
    const int n0   = blockIdx.x * BN;
    const int lane = tid & 31;
    const int wave = tid >> 5;   // 8 waves
    const int wm   = wave >> 1;  // 0..3  (M)
    const int wn   = wave & 1;   // 0..1  (N)

    v8f acc[2][2] = {};

    const int KTOT = (MODE == 0) ? 1152 : 2432;
    const int KW   = (MODE == 0) ? 1152 : 2304;

    // A staging: thread -> row m0+(tid>>1), k-range (tid&1)*8 .. +8
    const int a_m   = tid >> 1;
    const int a_k0  = (tid & 1) * 8;
    const int gm    = m0 + a_m;
    const int n_img = gm / 784;
    const int pq    = gm % 784;
    const int p     = pq / 28;
    const int q     = pq % 28;

    // B staging: thread -> col n0+(tid>>2), k-range (tid&3)*4 .. +4
    const int b_n  = tid >> 2;
    const int b_k0 = (tid & 3) * 4;
    const int gn   = n0 + b_n;

    for (int kt = 0; kt < KTOT; kt += BK) {
        if (kt + BK < KW)  // prefetch next weight tile (L2-resident stream)
            __builtin_prefetch(&w[gn * KW + kt + BK], 0, 0);

        // ---- stage B tile (weights): contiguous 16B per lane ----
        {
            const float* gsrc;
            if (MODE == 0)            gsrc = w + gn * 1152 + kt + b_k0;
            else if (kt < 2304)       gsrc = w + gn * 2304 + kt + b_k0;
            else                      gsrc = wid + gn * 128 + (kt - 2304) + b_k0;
#if HAVE_ASYNC_LDS
            __builtin_amdgcn_global_load_async_to_lds_b128(
                (gv4i_ptr)gsrc, (lv4i_ptr)&Bs[b_n][b_k0], 0, 0);
#else
#pragma unroll
            for (int j = 0; j < 4; ++j) Bs[b_n][b_k0 + j] = gsrc[j];
#endif
        }

        // ---- stage A tile (im2col gather, predicated zero-fill) ----
#pragma unroll
        for (int j = 0; j < 8; ++j) {
            const int kk = kt + a_k0 + j;
            float v = 0.0f;
            if (MODE == 0) {
                const int c = kk / 9, rs = kk % 9, r = rs / 3, s = rs % 3;
                const int h = 2 * p - 1 + r, ww = 2 * q - 1 + s;
                if ((unsigned)h < 56u && (unsigned)ww < 56u)
                    v = x[((n_img * 128 + c) * 56 + h) * 56 + ww];
            } else {
                if (kk < 2304) {
                    const int c = kk / 9, rs = kk % 9, r = rs / 3, s = rs % 3;
                    const int h = p - 1 + r, ww = q - 1 + s;
                    if ((unsigned)h < 28u && (unsigned)ww < 28u) {
                        const float t = act[((n_img * 256 + c) * 28 + h) * 28 + ww];
                        v = fminf(fmaxf(t, 0.0f), 1.0f);  // fused clamp01
                    }
                } else {  // residual 1x1 stride-2 path
                    const int c = kk - 2304;
                    v = x[((n_img * 128 + c) * 56 + 2 * p) * 56 + 2 * q];
                }
            }
            As[a_m][a_k0 + j] = v;
        }

#if HAVE_ASYNC_LDS
        __builtin_amdgcn_s_wait_asynccnt(0);  // our async LDS writes landed
#endif
        __syncthreads();

        // ---- 4 WMMA K-steps of 4, 2x2 tiles per wave ----
        const int am = wm * 32 + (lane & 15);
        const int bn = wn * 32 + (lane & 15);
        const int kl = (lane >> 4) * 2;  // f32 16x16x4 A/B lane K-slot
#pragma unroll
        for (int ks = 0; ks < BK; ks += 4) {
            v2f af[2], bf[2];
#pragma unroll
            for (int mt = 0; mt < 2; ++mt)
                af[mt] = *(const v2f*)&As[am + mt * 16][ks + kl];
#pragma unroll
            for (int nt = 0; nt < 2; ++nt)
                bf[nt] = *(const v2f*)&Bs[bn + nt * 16][ks + kl];
#pragma unroll
            for (int mt = 0; mt < 2; ++mt)
#pragma unroll
                for (int nt = 0; nt < 2; ++nt)
                    acc[mt][nt] = __builtin_amdgcn_wmma_f32_16x16x4_f32(
                        false, af[mt], false, bf[nt], (short)0, acc[mt][nt],
                        false, false);
        }
        __syncthreads();
    }

    // ---- store raw accumulators, NCHW ----
#pragma unroll
    for (int mt = 0; mt < 2; ++mt) {
#pragma unroll
        for (int nt = 0; nt < 2; ++nt) {
            const int colk = n0 + wn * 32 + nt * 16 + (lane & 15);
            const int mrow = m0 + wm * 32 + mt * 16 + (lane >> 4) * 8;
#pragma unroll
            for (int j = 0; j < 8; ++j) {
                const int g2  = mrow + j;
                const int ni  = g2 / 784;
                const int pq2 = g2 % 784;
                out[(ni * 256 + colk) * 784 + pq2] = acc[mt][nt][j];
            }
        }
    }
}

// ---------------------------------------------------------------------------
// Exact percentiles 1..99 via radix-histogram descent on monotonic keys.
// ---------------------------------------------------------------------------
struct Meta {
    int      slot_of_bin[4096];
    unsigned bin1[99];
    unsigned r1[99];
    int      slot1[99];
    unsigned prefix24[99];
    unsigned r2[99];
    int      slot2[99];
    unsigned plist[99];
    int      nprefix;
    int      nslots;
};

__device__ __forceinline__ unsigned key_of(float f) {
    unsigned u = __float_as_uint(f);
    return (u & 0x80000000u) ? ~u : (u ^ 0x80000000u);
}
__device__ __forceinline__ float val_of(unsigned k) {
    unsigned u = (k & 0x80000000u) ? (k ^ 0x80000000u) : ~k;
    return __uint_as_float(u);
}
// torch/py: k = 1 + round(0.01*q*(n-1)); zero-based rank = rint (half-to-even)
__device__ __forceinline__ unsigned long long rank_of(int q0, long long n) {
    double v = 0.01 * (double)(q0 + 1) * (double)(n - 1);
    return (unsigned long long)(long long)__builtin_rint(v);
}

__global__ void zero_u32(unsigned* p, int n) {
    int i = blockIdx.x * blockDim.x + threadIdx.x;
    if (i < n) p[i] = 0u;
}

__global__ void hist_l1(const float* __restrict__ src, int n, unsigned* __restrict__ h) {
    const int stride = gridDim.x * blockDim.x;
    for (int i = blockIdx.x * blockDim.x + threadIdx.x; i < n; i += stride)
        atomicAdd(&h[key_of(src[i]) >> 20], 1u);
}

__global__ void resolve_l1(const unsigned* __restrict__ h, Meta* mt, long long n) {
    for (int i = threadIdx.x; i < 4096; i += blockDim.x) mt->slot_of_bin[i] = -1;
    __syncthreads();
    if (threadIdx.x != 0) return;
    unsigned long long cum = 0;
    int qi = 0, ns = 0;
    for (int b = 0; b < 4096 && qi < 99; ++b) {
        const unsigned long long nxt = cum + h[b];
        bool first = true;
        while (qi < 99 && rank_of(qi, n) < nxt) {
            if (first) { mt->slot_of_bin[b] = ns++; first = false; }
            mt->bin1[qi]  = (unsigned)b;
            mt->slot1[qi] = ns - 1;
            mt->r1[qi]    = (unsigned)(rank_of(qi, n) - cum);
            ++qi;
        }
        cum = nxt;
    }
    mt->nslots = ns;
}

__global__ void hist_l2(const float* __restrict__ src, int n,
                        const Meta* __restrict__ mt, unsigned* __restrict__ h2) {
    __shared__ short sob[4096];
    for (int i = threadIdx.x; i < 4096; i += blockDim.x)
        sob[i] = (short)mt->slot_of_bin[i];
    __syncthreads();
    const int stride = gridDim.x * blockDim.x;
    for (int i = blockIdx.x * blockDim.x + threadIdx.x; i < n; i += stride) {
        const unsigned k = key_of(src[i]);
        const int s = sob[k >> 20];
        if (s >= 0) atomicAdd(&h2[s * 4096 + ((k >> 8) & 0xFFFu)], 1u);
    }
}

__global__ void resolve_l2(const unsigned* __restrict__ h2, Meta* mt) {
    const int q = threadIdx.x;
    if (q < 99) {
        const unsigned* h = h2 + mt->slot1[q] * 4096;
        const unsigned r = mt->r1[q];
        unsigned cum = 0;
        for (int b = 0; b < 4096; ++b) {
            const unsigned c = h[b];
            if (r < cum + c) {
                mt->prefix24[q] = (mt->bin1[q] << 12) | (unsigned)b;
                mt->r2[q] = r - cum;
                break;
            }
            cum += c;
        }
    }
    __syncthreads();
    if (threadIdx.x != 0) return;
    int np = 0;
    unsigned last = 0xFFFFFFFFu;
    for (int q2 = 0; q2 < 99; ++q2) {  // prefixes non-decreasing in q
        const unsigned pv = mt->prefix24[q2];
        if (np == 0 || pv != last) { mt->plist[np++] = pv; last = pv; }
        mt->slot2[q2] = np - 1;
    }
    mt->nprefix = np;
}

__global__ void hist_l3(const float* __restrict__ src, int n,
                        const Meta* __restrict__ mt, unsigned* __restrict__ h3) {
    __shared__ unsigned pl[99];
    __shared__ int npv;
    if (threadIdx.x == 0) npv = mt->nprefix;
    for (int i = threadIdx.x; i < 99; i += blockDim.x) pl[i] = mt->plist[i];
    __syncthreads();
    const int np = npv;
    const int stride = gridDim.x * blockDim.x;
    for (int i = blockIdx.x * blockDim.x + threadIdx.x; i < n; i += stride) {
        const unsigned k = key_of(src[i]);
        const unsigned pfx = k >> 8;
        int lo = 0, hi = np;  // binary search (pl sorted ascending)
        while (lo < hi) {
            const int mid = (lo + hi) >> 1;
            if (pl[mid] < pfx) lo = mid + 1; else hi = mid;
        }
        if (lo < np && pl[lo] == pfx) atomicAdd(&h3[lo * 256 + (k & 255u)], 1u);
    }
}

__global__ void resolve_l3(const unsigned* __restrict__ h3, const Meta* __restrict__ mt,
                           float* __restrict__ dst) {
    const int q = threadIdx.x;
    if (q >= 99) return;
    const unsigned* h = h3 + mt->slot2[q] * 256;
    const unsigned r = mt->r2[q];
    unsigned cum = 0, key = 0;
    for (int t = 0; t < 256; ++t) {
        const unsigned c = h[t];
        if (r < cum + c) { key = (mt->prefix24[q] << 8) | (unsigned)t; break; }
        cum += c;
    }
    dst[q] = val_of(key);
}

__global__ void clamp01(float* __restrict__ p, int n) {
    const int stride = gridDim.x * blockDim.x;
    for (int i = blockIdx.x * blockDim.x + threadIdx.x; i < n; i += stride)
        p[i] = fminf(fmaxf(p[i], 0.0f), 1.0f);
}

// ---------------------------------------------------------------------------
extern "C" void kernel_launch(void* const* d_in, const int* in_sizes, int n_in,
                              void* d_out, int out_size, void* d_ws, size_t ws_size,
                              hipStream_t stream) {
    const float* x   = (const float*)d_in[0];
    const float* w1  = (const float*)d_in[1];
    const float* w2  = (const float*)d_in[2];
    const float* wid = (const float*)d_in[3];
    float* out = (float*)d_out;
    const int nelts = (int)NOUT_LL;

    // workspace: raw conv1 output + percentile scratch (~53 MB total)
    char* ws = (char*)d_ws;
    float* out1 = (float*)ws;
    size_t off = ((size_t)NOUT_LL * 4 + 255) & ~(size_t)255;
    Meta* mt = (Meta*)(ws + off);
    off = (off + sizeof(Meta) + 255) & ~(size_t)255;
    unsigned* hist1 = (unsigned*)(ws + off); off += 4096ull * 4;
    unsigned* hist2 = (unsigned*)(ws + off); off += 99ull * 4096 * 4;
    unsigned* hist3 = (unsigned*)(ws + off);

    const dim3 grid_conv(4, 392);  // N tiles (256/64) x M tiles (50176/128)

    // conv1 -> raw out1 (in ws)
    conv_gemm<0><<<grid_conv, NTHREADS, 0, stream>>>(x, w1, nullptr, nullptr, out1);

    auto percentiles = [&](const float* src, float* dst) {
        zero_u32<<<(4096 + 255) / 256, 256, 0, stream>>>(hist1, 4096);
        hist_l1<<<2048, 256, 0, stream>>>(src, nelts, hist1);
        resolve_l1<<<1, 256, 0, stream>>>(hist1, mt, NOUT_LL);
        zero_u32<<<(99 * 4096 + 255) / 256, 256, 0, stream>>>(hist2, 99 * 4096);
        hist_l2<<<2048, 256, 0, stream>>>(src, nelts, mt, hist2);
        resolve_l2<<<1, 128, 0, stream>>>(hist2, mt);
        zero_u32<<<(99 * 256 + 255) / 256, 256, 0, stream>>>(hist3, 99 * 256);
        hist_l3<<<2048, 256, 0, stream>>>(src, nelts, mt, hist3);
        resolve_l3<<<1, 128, 0, stream>>>(hist3, mt, dst);
    };

    // a = percentiles of raw conv1 output
    percentiles(out1, out + NOUT_LL);

    // conv2(clamp01(out1)) + 1x1/s2 residual -> raw sum directly into d_out
    conv_gemm<1><<<grid_conv, NTHREADS, 0, stream>>>(x, w2, wid, out1, out);

    // b = percentiles of raw sum
    percentiles(out, out + NOUT_LL + 99);

    // final clamp in place
    clamp01<<<2048, 256, 0, stream>>>(out, nelts);
}